// first_order_low_pass_layer_14946486190127
// MI455X (gfx1250) — compile-verified
//
#include <hip/hip_runtime.h>

typedef float v4f __attribute__((ext_vector_type(4)));
typedef int v4i __attribute__((__vector_size__(4 * sizeof(int))));
typedef __attribute__((address_space(1))) v4i as1_v4i;  // global
typedef __attribute__((address_space(3))) v4i as3_v4i;  // LDS

constexpr int kB = 32;
constexpr int kD = 2048;
constexpr int kT = 512;
constexpr int kThreads = 256;
constexpr int kChunk = 32;               // floats staged per thread per stage
constexpr int kStages = kT / kChunk;     // 16
constexpr int kLdsStride = kChunk + 4;   // 36 floats = 144B: 16B-aligned, bank-skewed
constexpr int kBufFloats = kThreads * kLdsStride;
constexpr int kCpolNT = 1;               // gfx12+ CPol: TH[2:0] = 1 (non-temporal)

// One async DMA of 16B/lane from global to this lane's private LDS slice.
__device__ __forceinline__ void async_copy16(const float* gsrc, float* ldst) {
#if __has_builtin(__builtin_amdgcn_global_load_async_to_lds_b128)
  // Low 32 bits of a flat LDS address are the AS(3) offset on AMDGPU; the
  // global aperture is identity, so integer casts recover the typed pointers.
  __builtin_amdgcn_global_load_async_to_lds_b128(
      (as1_v4i*)(unsigned long long)(const void*)gsrc,
      (as3_v4i*)(unsigned int)(unsigned long long)(void*)ldst,
      /*offset=*/0, /*cpol=*/kCpolNT);
#else
  *(v4f*)ldst = *(const v4f*)gsrc;  // synchronous fallback (also host pass)
#endif
}

template <int N>
__device__ __forceinline__ void async_wait() {
#if __has_builtin(__builtin_amdgcn_s_wait_asynccnt)
  __builtin_amdgcn_s_wait_asynccnt(N);
#elif defined(__AMDGCN__)
  asm volatile("s_wait_asynccnt %0" ::"n"(N) : "memory");
#endif
}

__global__ __launch_bounds__(kThreads) void ema_scan_kernel(
    const float* __restrict__ x, const float* __restrict__ alpha,
    float* __restrict__ psp, float* __restrict__ fin) {
  __shared__ float lds[2 * kBufFloats];

  const int tid = threadIdx.x;
  const int row = blockIdx.x * kThreads + tid;  // row = b * D + d
  const int d = row & (kD - 1);
  const float a = alpha[d];

  const float* grow = x + (unsigned long long)row * kT;
  float* orow = psp + (unsigned long long)row * kT;

  // Prologue: stage 0 -> buffer 0. Each lane fills only its own LDS slice,
  // so the wave-scoped s_wait_asynccnt is the only sync needed (no barriers).
  {
    float* l = &lds[tid * kLdsStride];
#pragma unroll
    for (int k = 0; k < kChunk / 4; ++k) async_copy16(grow + 4 * k, l + 4 * k);
  }

  float s = 0.0f;
  for (int stage = 0; stage < kStages; ++stage) {
    float* cur = &lds[(stage & 1) * kBufFloats + tid * kLdsStride];

    if (stage + 1 < kStages) {
      // Kick off next stage, then wait for the current one (8 still in flight).
      const float* g = grow + (stage + 1) * kChunk;
      float* nxt = &lds[((stage + 1) & 1) * kBufFloats + tid * kLdsStride];
#pragma unroll
      for (int k = 0; k < kChunk / 4; ++k) async_copy16(g + 4 * k, nxt + 4 * k);
      async_wait<kChunk / 4>();
    } else {
      async_wait<0>();
    }

    float* gout = orow + stage * kChunk;
#pragma unroll
    for (int k = 0; k < kChunk / 4; ++k) {
      v4f v = *(const v4f*)(cur + 4 * k);  // ds_load_b128
      v4f o;
      s = __builtin_fmaf(a, s, v.x); o.x = s;
      s = __builtin_fmaf(a, s, v.y); o.y = s;
      s = __builtin_fmaf(a, s, v.z); o.z = s;
      s = __builtin_fmaf(a, s, v.w); o.w = s;
      // Output is streamed once and never re-read: non-temporal b128 store.
      __builtin_nontemporal_store(o, (v4f*)(gout + 4 * k));
    }
  }
  fin[row] = s;
}

extern "C" void kernel_launch(void* const* d_in, const int* in_sizes, int n_in,
                              void* d_out, int out_size, void* d_ws, size_t ws_size,
                              hipStream_t stream) {
  (void)in_sizes; (void)n_in; (void)out_size; (void)d_ws; (void)ws_size;
  const float* x = (const float*)d_in[0];      // input_spikes [B, D, T] f32
  const float* alpha = (const float*)d_in[1];  // alpha [D] f32
  float* psp = (float*)d_out;                                  // [B, D, T]
  float* fin = psp + (unsigned long long)kB * kD * kT;         // [B, D]
  const int rows = kB * kD;  // 65536 rows -> 256 blocks of 256 threads
  ema_scan_kernel<<<rows / kThreads, kThreads, 0, stream>>>(x, alpha, psp, fin);
}